// OwlViTPredictionHead_6820408066753
// MI455X (gfx1250) — compile-verified
//
#include <hip/hip_runtime.h>
#include <hip/hip_bf16.h>
#include <math.h>

// ---------------------------------------------------------------------------
// OwlViT prediction head for MI455X (gfx1250, wave32, WMMA + async-to-LDS).
// Dominant cost: reading query_embeds (629 MB) once -> ~27us @ 23.3 TB/s.
// logits kernel: double-buffered GLOBAL_LOAD_ASYNC_TO_LDS_B128 staging of q
// (ASYNCcnt-pipelined) feeding V_WMMA_F32_16X16X4_F32, with the q-row L2
// norm fused out of the B fragments (q is read from HBM exactly once).
// ---------------------------------------------------------------------------

typedef __attribute__((ext_vector_type(2))) float v2f;
typedef __attribute__((ext_vector_type(8))) float v8f;

static constexpr int Bc  = 32;         // batch
static constexpr int Pc  = 576;        // patches
static constexpr int DVc = 768;        // vision dim
static constexpr int DTc = 512;        // text dim
static constexpr int NPc = 12;         // parts
static constexpr int NQc = 800 * NPc;  // 9600 query rows per batch
static constexpr int Mrows = Bc * NPc; // 384

static constexpr int QLDS = 68;        // padded LDS row stride (floats): 16B
                                       // aligned, stride%64==4 -> conflict-free
static constexpr int ALDS = 520;       // img LDS row stride (floats)

__device__ __forceinline__ v8f wmma_f32_16x16x4(v2f a, v2f b, v8f c) {
  // D = A(16x4,f32) * B(4x16,f32) + C(16x16,f32)
  return __builtin_amdgcn_wmma_f32_16x16x4_f32(false, a, false, b,
                                               (short)0, c, false, false);
}

__device__ __forceinline__ float gelu_erf(float x) {
  return 0.5f * x * (1.0f + erff(x * 0.70710678118654752f));
}

// ---------------------------------------------------------------------------
// K1: dedup'd top-k gather:  parts[b,n,:] = sum_unique image[b, idx[b,n,k], :]
// topk layout: [B, K=3, N_PARTS=12] int64; reference scatter SET collapses dups.
// ---------------------------------------------------------------------------
__global__ void gather_parts_kernel(const long long* __restrict__ topk,
                                    const float* __restrict__ image,
                                    float* __restrict__ parts) {
  const int bn = blockIdx.x;            // 0..383
  const int b = bn / NPc, n = bn % NPc;
  const long long i0 = topk[b * 3 * NPc + 0 * NPc + n];
  const long long i1 = topk[b * 3 * NPc + 1 * NPc + n];
  const long long i2 = topk[b * 3 * NPc + 2 * NPc + n];
  const bool u1 = (i1 != i0);
  const bool u2 = (i2 != i0) && (i2 != i1);
  const float* base = image + (size_t)b * Pc * DVc;
  for (int d = threadIdx.x; d < DVc; d += blockDim.x) {
    float s = base[(size_t)i0 * DVc + d];
    if (u1) s += base[(size_t)i1 * DVc + d];
    if (u2) s += base[(size_t)i2 * DVc + d];
    parts[(size_t)bn * DVc + d] = s;
  }
}

// ---------------------------------------------------------------------------
// K2-K4: Y = gelu(X @ W^T + bias).  X:[M,K] row-major, W:[N,K] row-major.
// Block = 256 thr = 8 waves; each wave owns one 16x16 output tile.
// grid = (M/16, N/128).
// ---------------------------------------------------------------------------
__global__ void gemm_gelu_wmma_kernel(const float* __restrict__ X,
                                      const float* __restrict__ W,
                                      const float* __restrict__ bias,
                                      float* __restrict__ Y,
                                      int M, int N, int K) {
  const int wave = threadIdx.x >> 5;
  const int lane = threadIdx.x & 31;
  const int half = lane >> 4;   // which K pair / which M half of C
  const int l16  = lane & 15;
  const int m0 = blockIdx.x * 16;
  const int n0 = blockIdx.y * 128 + wave * 16;

  const float* Xrow = X + (size_t)(m0 + l16) * K;  // A: lane row (ISA 16x4 layout)
  const float* Wrow = W + (size_t)(n0 + l16) * K;  // B column n = W row n

  v8f c = {};
  for (int k = 0; k < K; k += 4) {
    const int ka = k + half * 2;
    v2f a, b;
    a.x = Xrow[ka];     a.y = Xrow[ka + 1];
    b.x = Wrow[ka];     b.y = Wrow[ka + 1];
    c = wmma_f32_16x16x4(a, b, c);
  }

  const float bcol = bias[n0 + l16];
#pragma unroll
  for (int r = 0; r < 8; ++r) {
    const int m = m0 + half * 8 + r;       // C/D layout: lanes>=16 hold M=8..15
    float v = gelu_erf(c[r] + bcol);
    if (m < M) Y[(size_t)m * N + (n0 + l16)] = v;
  }
}

// ---------------------------------------------------------------------------
// K5: in-place row L2 normalization (img: 384 rows x 512).
// ---------------------------------------------------------------------------
__global__ void l2norm_rows_kernel(float* __restrict__ x, int ncols) {
  __shared__ float red[8];
  float* p = x + (size_t)blockIdx.x * ncols;
  float s = 0.0f;
  for (int d = threadIdx.x; d < ncols; d += blockDim.x) {
    const float v = p[d];
    s += v * v;
  }
#pragma unroll
  for (int off = 16; off > 0; off >>= 1) s += __shfl_xor(s, off, 32);
  if ((threadIdx.x & 31) == 0) red[threadIdx.x >> 5] = s;
  __syncthreads();
  if (threadIdx.x == 0) {
    float t = 0.0f;
#pragma unroll
    for (int w = 0; w < 8; ++w) t += red[w];
    red[0] = 1.0f / fmaxf(sqrtf(t), 1e-12f);
  }
  __syncthreads();
  const float inv = red[0];
  for (int d = threadIdx.x; d < ncols; d += blockDim.x) p[d] *= inv;
}

// ---------------------------------------------------------------------------
// Async staging of one q chunk: 16 rows x 64 floats -> wave-private LDS
// (padded row stride QLDS). 8 x GLOBAL_LOAD_ASYNC_TO_LDS_B128 per wave
// (16B/lane/issue); tracked by ASYNCcnt, no VGPR data movement.
// ---------------------------------------------------------------------------
__device__ __forceinline__ void stage_q_async(float* __restrict__ buf,
                                              const float* __restrict__ gsrc,
                                              int lane) {
#pragma unroll
  for (int it = 0; it < 8; ++it) {
    const int t   = it * 32 + lane;     // float4 index within 16x64 tile
    const int row = t >> 4;             // 16 float4 per row
    const int c4  = t & 15;
    const float*   g   = gsrc + (size_t)row * DTc + c4 * 4;
    const unsigned lds = (unsigned)(uintptr_t)(buf + row * QLDS + c4 * 4);
    asm volatile("global_load_async_to_lds_b128 %0, %1, off"
                 :: "v"(lds), "v"(g) : "memory");
  }
}

__device__ __forceinline__ void compute_chunk(const float* __restrict__ As,
                                              const float* __restrict__ buf,
                                              int k0, int l16, int half,
                                              float& sq, v8f& c) {
#pragma unroll
  for (int kk = 0; kk < 64; kk += 4) {
    const int ka = kk + half * 2;
    v2f a, bq;
    a.x  = As[l16 * ALDS + k0 + ka];
    a.y  = As[l16 * ALDS + k0 + ka + 1];
    bq.x = buf[l16 * QLDS + ka];
    bq.y = buf[l16 * QLDS + ka + 1];
    sq += bq.x * bq.x + bq.y * bq.y;    // free q-norm accumulation
    c = wmma_f32_16x16x4(a, bq, c);
  }
}

// ---------------------------------------------------------------------------
// K6: fused logits GEMM + q L2-norm.
//   logits[b, m, i] = img_n[b,m,:] . q[b,i,:] / max(||q[b,i,:]||, eps)
// Block: b = blockIdx.x, 8 waves; wave handles q-row tile i0 = by*128+wave*16.
// q double-buffered in LDS via async loads: issue chunk ch+1, wait
// ASYNCcnt<=8 (async loads complete in order => chunk ch landed), consume.
// Row sum-of-squares accumulated from B fragments (each q element appears in
// exactly one lane/frag slot), combined across halves with __shfl_xor(16).
// ---------------------------------------------------------------------------
__global__ void logits_fused_kernel(const float* __restrict__ img,   // [B*12,512] normalized
                                    const float* __restrict__ query, // [B,9600,512]
                                    float* __restrict__ logits) {    // [B*12,9600]
  __shared__ float As[16 * ALDS];
  __shared__ float Qs[8][2][16 * QLDS];

  const int b    = blockIdx.x;
  const int wave = threadIdx.x >> 5;
  const int lane = threadIdx.x & 31;
  const int half = lane >> 4;
  const int l16  = lane & 15;
  const int i0   = blockIdx.y * 128 + wave * 16;

  // Stage img tile (float4 / b128), zero-padding rows 12..15 so WMMA runs
  // with EXEC all-1s.
  for (int e = threadIdx.x; e < 16 * (DTc / 4); e += blockDim.x) {
    const int row = e >> 7;             // 128 float4 per row
    const int c4  = e & 127;
    float4 v = make_float4(0.0f, 0.0f, 0.0f, 0.0f);
    if (row < NPc)
      v = *(const float4*)&img[((size_t)b * NPc + row) * DTc + c4 * 4];
    *(float4*)&As[row * ALDS + c4 * 4] = v;
  }
  __syncthreads();

  const float* qb = query + ((size_t)b * NQc + i0) * DTc;
  float* buf0 = &Qs[wave][0][0];
  float* buf1 = &Qs[wave][1][0];

  float sq = 0.0f;
  v8f c = {};

  stage_q_async(buf0, qb, lane);                       // chunk 0
#pragma unroll
  for (int ch = 0; ch < 7; ++ch) {
    float*       nbuf = (ch & 1) ? buf0 : buf1;
    const float* cbuf = (ch & 1) ? buf1 : buf0;
    stage_q_async(nbuf, qb + (ch + 1) * 64, lane);     // prefetch chunk ch+1
    asm volatile("s_wait_asynccnt 0x8" ::: "memory");  // chunk ch has landed
    compute_chunk(As, cbuf, ch * 64, l16, half, sq, c);
  }
  asm volatile("s_wait_asynccnt 0x0" ::: "memory");    // final chunk landed
  compute_chunk(As, buf1, 7 * 64, l16, half, sq, c);

  sq += __shfl_xor(sq, 16, 32);          // combine K%4 {0,1} half with {2,3} half
  const float inv = 1.0f / fmaxf(sqrtf(sq), 1e-12f);

#pragma unroll
  for (int r = 0; r < 8; ++r) {
    const int m = half * 8 + r;
    if (m < NPc)
      logits[((size_t)b * NPc + m) * NQc + (i0 + l16)] = c[r] * inv;
  }
}

// ---------------------------------------------------------------------------
// K7: pred_logits[b,c] = sum_n logits[b, n, c*12 + n]
// ---------------------------------------------------------------------------
__global__ void pred_logits_kernel(const float* __restrict__ logits,
                                   float* __restrict__ pred) {
  const int t = blockIdx.x * blockDim.x + threadIdx.x;
  if (t >= Bc * 800) return;
  const int b = t / 800, ccls = t % 800;
  const float* lb = logits + (size_t)b * NPc * NQc;
  float s = 0.0f;
#pragma unroll
  for (int n = 0; n < NPc; ++n) s += lb[(size_t)n * NQc + ccls * NPc + n];
  pred[t] = s;
}

// ---------------------------------------------------------------------------
extern "C" void kernel_launch(void* const* d_in, const int* in_sizes, int n_in,
                              void* d_out, int out_size, void* d_ws, size_t ws_size,
                              hipStream_t stream) {
  const float*     image = (const float*)d_in[0];
  const float*     query = (const float*)d_in[1];
  const long long* topk  = (const long long*)d_in[2];
  const float*     W1    = (const float*)d_in[3];
  const float*     b1    = (const float*)d_in[4];
  const float*     W2    = (const float*)d_in[5];
  const float*     b2    = (const float*)d_in[6];
  const float*     W3    = (const float*)d_in[7];
  const float*     b3    = (const float*)d_in[8];

  float* logits = (float*)d_out;                 // [384, 9600]
  float* pred   = logits + (size_t)Mrows * NQc;  // [32, 800]

  float* parts = (float*)d_ws;                   // [384, 768]
  float* h1    = parts + (size_t)Mrows * DVc;    // [384, 768]
  float* h2    = h1    + (size_t)Mrows * DVc;    // [384, 768]
  float* imgn  = h2    + (size_t)Mrows * DVc;    // [384, 512]

  gather_parts_kernel<<<Mrows, 256, 0, stream>>>(topk, image, parts);

  dim3 gA(Mrows / 16, DVc / 128);   // (24, 6)
  gemm_gelu_wmma_kernel<<<gA, 256, 0, stream>>>(parts, W1, b1, h1, Mrows, DVc, DVc);
  gemm_gelu_wmma_kernel<<<gA, 256, 0, stream>>>(h1,    W2, b2, h2, Mrows, DVc, DVc);

  dim3 gC(Mrows / 16, DTc / 128);   // (24, 4)
  gemm_gelu_wmma_kernel<<<gC, 256, 0, stream>>>(h2,    W3, b3, imgn, Mrows, DTc, DVc);

  l2norm_rows_kernel<<<Mrows, 256, 0, stream>>>(imgn, DTc);

  dim3 gL(Bc, NQc / 128);           // (32, 75)
  logits_fused_kernel<<<gL, 256, 0, stream>>>(imgn, query, logits);

  pred_logits_kernel<<<(Bc * 800 + 255) / 256, 256, 0, stream>>>(logits, pred);
}